// YOWO_19447611916925
// MI455X (gfx1250) — compile-verified
//
#include <hip/hip_runtime.h>
#include <math.h>

typedef __attribute__((ext_vector_type(2))) float v2f;
typedef __attribute__((ext_vector_type(8))) float v8f;

#define NB 16          // batches
#define TOT 16464      // 112*112 + 56*56 + 28*28
#define KSEL 500
#define KALL 1500
#define CONF_T 0.3f
#define NMS_T 0.5f
#define IMG_INV (1.0f / 896.0f)

// -------------------------------------------------------------------------
// Kernel 1: fused 1x1-conv heads via V_WMMA_F32_16X16X4_F32.
// D[16 pos x 16 out] = A[16 pos x K=512] * B[512 x 16].
// K 0..255   -> reg_feat channels, outputs n=0 (conf), n=1..4 (reg)
// K 256..511 -> cls_feat channels, output  n=5 (cls)
// MLVL compile-time => channel stride is an immediate; pointers are simple
// induction variables; tail handled by clamping (no exec divergence).
// -------------------------------------------------------------------------
template <int MLVL, int LVL_OFF>
__global__ __launch_bounds__(128) void head_wmma_kernel(
    const float* __restrict__ regf, const float* __restrict__ clsf,
    const float* __restrict__ Wconf, const float* __restrict__ Wcls,
    const float* __restrict__ Wreg, float* __restrict__ head) {
  __shared__ float wlds[512 * 16];  // 32 KB padded weight tile, k-major

  // Stage B matrix (weights) into LDS, zero-padded to 16 output columns.
  for (int i = threadIdx.x; i < 512 * 16; i += blockDim.x) {
    int k = i >> 4, n = i & 15;
    float v = 0.f;
    if (k < 256) {
      if (n == 0) v = Wconf[k];
      else if (n >= 1 && n <= 4) v = Wreg[(n - 1) * 256 + k];
    } else if (n == 5) {
      v = Wcls[k - 256];
    }
    wlds[i] = v;
  }
  __syncthreads();

  const int lane = threadIdx.x & 31;
  const int wave = threadIdx.x >> 5;
  const int tile = blockIdx.x * 4 + wave;
  const int b = blockIdx.y;
  const int m0 = tile * 16;
  if (m0 >= MLVL) return;  // wave-uniform exit

  const int half = lane >> 4;  // 0: K pair {0,1}, 1: K pair {2,3}
  const int nl = lane & 15;    // A: row (position); B/D: column (output)
  int m = m0 + nl;
  if (m >= MLVL) m = MLVL - 1;  // clamp tail: always in-bounds, no masking

  const size_t cs = (size_t)MLVL;  // compile-time channel stride
  const float* pr = regf + ((size_t)b * 256 + 2 * half) * cs + m;
  const float* pc = clsf + ((size_t)b * 256 + 2 * half) * cs + m;
  const int wbase = half * 32 + nl;  // LDS idx of weight row (k0+2*half), col nl

  v8f acc = {0.f, 0.f, 0.f, 0.f, 0.f, 0.f, 0.f, 0.f};

#pragma unroll 8
  for (int k0 = 0; k0 < 256; k0 += 4) {
    v2f av = {__builtin_nontemporal_load(pr),
              __builtin_nontemporal_load(pr + cs)};
    v2f bv = {wlds[k0 * 16 + wbase], wlds[k0 * 16 + 16 + wbase]};
    pr += 4 * cs;
    acc = __builtin_amdgcn_wmma_f32_16x16x4_f32(false, av, false, bv,
                                                (short)0, acc, false, false);
  }
#pragma unroll 8
  for (int k0 = 256; k0 < 512; k0 += 4) {
    v2f av = {__builtin_nontemporal_load(pc),
              __builtin_nontemporal_load(pc + cs)};
    v2f bv = {wlds[k0 * 16 + wbase], wlds[k0 * 16 + 16 + wbase]};
    pc += 4 * cs;
    acc = __builtin_amdgcn_wmma_f32_16x16x4_f32(false, av, false, bv,
                                                (short)0, acc, false, false);
  }

  // D layout: VGPR j holds row M = j + 8*half, column N = nl.
  if (nl < 6) {
    const int mb = half * 8;
#pragma unroll
    for (int j = 0; j < 8; ++j) {
      int mm = m0 + mb + j;
      if (mm < MLVL)
        head[((size_t)b * TOT + LVL_OFF + mm) * 6 + nl] = acc[j];
    }
  }
}

// -------------------------------------------------------------------------
// Kernel 2: fused score = sqrt(sigmoid(conf+b) * sigmoid(cls+b))
// -------------------------------------------------------------------------
__device__ __forceinline__ float sigm(float x) { return 1.f / (1.f + expf(-x)); }

__global__ void score_kernel(const float* __restrict__ head,
                             const float* __restrict__ bconf,
                             const float* __restrict__ bcls,
                             float* __restrict__ scores) {
  int gid = blockIdx.x * blockDim.x + threadIdx.x;
  if (gid >= NB * TOT) return;
  int r = gid % TOT;
  int l = (r < 12544) ? 0 : ((r < 15680) ? 1 : 2);
  const float* h = head + (size_t)gid * 6;
  float conf = h[0] + bconf[l];
  float cls = h[5] + bcls[l];
  scores[gid] = sqrtf(sigm(conf) * sigm(cls));
}

// -------------------------------------------------------------------------
// Kernel 3: exact top-500 per (batch, level) via 4-round radix select on
// positive-float bits; ties filled in ascending index order. Decodes boxes.
// -------------------------------------------------------------------------
__global__ __launch_bounds__(256) void topk_kernel(
    const float* __restrict__ scores, const float* __restrict__ head,
    const float* __restrict__ breg, float* __restrict__ sel_s,
    float* __restrict__ sel_b) {
  const int LM[3] = {12544, 3136, 784};
  const int LOFF[3] = {0, 12544, 15680};
  const int LW[3] = {112, 56, 28};
  const float LST[3] = {8.f, 16.f, 32.f};

  const int l = blockIdx.x;
  const int b = blockIdx.y;
  const int M = LM[l];
  const float* sc = scores + (size_t)b * TOT + LOFF[l];

  __shared__ unsigned hist[1024];
  __shared__ unsigned s_prefix, s_pmask, s_need;
  __shared__ int s_cnt;

  if (threadIdx.x == 0) { s_prefix = 0u; s_pmask = 0u; s_need = KSEL; }
  __syncthreads();

  const int shifts[4] = {22, 12, 2, 0};
  const unsigned bmasks[4] = {1023u, 1023u, 1023u, 3u};
  for (int rnd = 0; rnd < 4; ++rnd) {
    const int sh = shifts[rnd];
    const unsigned bm = bmasks[rnd];
    for (int i = threadIdx.x; i < 1024; i += blockDim.x) hist[i] = 0u;
    __syncthreads();
    unsigned pm = s_pmask, pf = s_prefix;
    for (int i = threadIdx.x; i < M; i += blockDim.x) {
      unsigned k = __float_as_uint(sc[i]);
      if ((k & pm) == pf) atomicAdd(&hist[(k >> sh) & bm], 1u);
    }
    __syncthreads();
    if (threadIdx.x == 0) {
      unsigned need = s_need, cum = 0u;
      int T = 0;
      for (int bkt = (int)bm; bkt >= 0; --bkt) {
        unsigned c = hist[bkt];
        if (cum + c >= need) { T = bkt; break; }
        cum += c;
      }
      s_need = need - cum;
      s_prefix = pf | ((unsigned)T << sh);
      s_pmask = pm | (bm << sh);
    }
    __syncthreads();
  }

  const unsigned thr = s_prefix;
  if (threadIdx.x == 0) s_cnt = 0;
  __syncthreads();

  auto emit = [&](int slot, int i) {
    int o = b * KALL + l * KSEL + slot;
    sel_s[o] = sc[i];
    const float* hv = head + ((size_t)b * TOT + LOFF[l] + i) * 6;
    float r0 = hv[1] + breg[l * 4 + 0];
    float r1 = hv[2] + breg[l * 4 + 1];
    float r2 = hv[3] + breg[l * 4 + 2];
    float r3 = hv[4] + breg[l * 4 + 3];
    float st = LST[l];
    int wq = i % LW[l], hq = i / LW[l];
    float cx = ((float)wq + 0.5f) * st + r0 * st;
    float cy = ((float)hq + 0.5f) * st + r1 * st;
    float hw = 0.5f * expf(r2) * st;
    float hh = 0.5f * expf(r3) * st;
    sel_b[(size_t)o * 4 + 0] = cx - hw;
    sel_b[(size_t)o * 4 + 1] = cy - hh;
    sel_b[(size_t)o * 4 + 2] = cx + hw;
    sel_b[(size_t)o * 4 + 3] = cy + hh;
  };

  for (int i = threadIdx.x; i < M; i += blockDim.x) {
    if (__float_as_uint(sc[i]) > thr) {
      int p = atomicAdd(&s_cnt, 1);
      emit(p, i);
    }
  }
  __syncthreads();
  if (threadIdx.x == 0) {  // index-ordered tie fill (rare path)
    int p = s_cnt;
    int rem = (int)s_need;
    for (int i = 0; i < M && rem > 0; ++i) {
      if (__float_as_uint(sc[i]) == thr) { emit(p++, i); --rem; }
    }
  }
}

// -------------------------------------------------------------------------
// Kernel 4: per-batch bitonic sort (2048) + greedy NMS + normalize + write.
// -------------------------------------------------------------------------
__global__ __launch_bounds__(1024) void sort_nms_kernel(
    const float* __restrict__ sel_s, const float* __restrict__ sel_b,
    float* __restrict__ out) {
  const int b = blockIdx.x;
  const int nt = blockDim.x;
  __shared__ float key[2048];
  __shared__ int pay[2048];
  __shared__ float bx[KALL * 4];
  __shared__ int keep[KALL];

  for (int i = threadIdx.x; i < 2048; i += nt) {
    if (i < KALL) { key[i] = -sel_s[b * KALL + i]; pay[i] = i; }
    else          { key[i] = 1e30f;                pay[i] = -1; }
  }
  __syncthreads();

  // ascending bitonic on -score == descending by score
  for (int k2 = 2; k2 <= 2048; k2 <<= 1) {
    for (int j = k2 >> 1; j > 0; j >>= 1) {
      __syncthreads();
      for (int i = threadIdx.x; i < 2048; i += nt) {
        int ixj = i ^ j;
        if (ixj > i) {
          bool up = ((i & k2) == 0);
          float a = key[i], c = key[ixj];
          if (up ? (a > c) : (a < c)) {
            key[i] = c; key[ixj] = a;
            int t = pay[i]; pay[i] = pay[ixj]; pay[ixj] = t;
          }
        }
      }
    }
  }
  __syncthreads();

  for (int r = threadIdx.x; r < KALL; r += nt) {
    float s = -key[r];
    key[r] = s;
    int si = pay[r];
    const float* sb = sel_b + ((size_t)b * KALL + si) * 4;
    bx[r * 4 + 0] = sb[0]; bx[r * 4 + 1] = sb[1];
    bx[r * 4 + 2] = sb[2]; bx[r * 4 + 3] = sb[3];
    keep[r] = (s > CONF_T) ? 1 : 0;
  }

  // greedy sequential NMS, parallel over j
  for (int i = 0; i < KALL; ++i) {
    __syncthreads();
    if (keep[i]) {
      float x1 = bx[i * 4 + 0], y1 = bx[i * 4 + 1];
      float x2 = bx[i * 4 + 2], y2 = bx[i * 4 + 3];
      float ai = fmaxf(x2 - x1, 0.f) * fmaxf(y2 - y1, 0.f);
      for (int j = i + 1 + threadIdx.x; j < KALL; j += nt) {
        if (!keep[j]) continue;
        float u1 = bx[j * 4 + 0], v1 = bx[j * 4 + 1];
        float u2 = bx[j * 4 + 2], v2 = bx[j * 4 + 3];
        float aj = fmaxf(u2 - u1, 0.f) * fmaxf(v2 - v1, 0.f);
        float ix = fminf(x2, u2) - fmaxf(x1, u1);
        float iy = fminf(y2, v2) - fmaxf(y1, v1);
        float inter = fmaxf(ix, 0.f) * fmaxf(iy, 0.f);
        float iou = inter / fmaxf(ai + aj - inter, 1e-9f);
        if (iou > NMS_T) keep[j] = 0;
      }
    }
  }
  __syncthreads();

  for (int r = threadIdx.x; r < KALL; r += nt) {
    float s = key[r];
    float kf = keep[r] ? 1.f : 0.f;
    float* o = out + ((size_t)b * KALL + r) * 6;
    o[0] = s * kf;
#pragma unroll
    for (int c = 0; c < 4; ++c)
      o[1 + c] = fminf(fmaxf(bx[r * 4 + c] * IMG_INV, 0.f), 1.f);
    o[5] = kf;
  }
}

// -------------------------------------------------------------------------
extern "C" void kernel_launch(void* const* d_in, const int* in_sizes, int n_in,
                              void* d_out, int out_size, void* d_ws,
                              size_t ws_size, hipStream_t stream) {
  const float* cls_feat[3] = {(const float*)d_in[0], (const float*)d_in[1],
                              (const float*)d_in[2]};
  const float* reg_feat[3] = {(const float*)d_in[3], (const float*)d_in[4],
                              (const float*)d_in[5]};
  const float* Wconf = (const float*)d_in[6];
  const float* bconf = (const float*)d_in[7];
  const float* Wcls = (const float*)d_in[8];
  const float* bcls = (const float*)d_in[9];
  const float* Wreg = (const float*)d_in[10];
  const float* breg = (const float*)d_in[11];

  float* ws = (float*)d_ws;
  float* head = ws;                                  // NB*TOT*6
  float* scores = head + (size_t)NB * TOT * 6;       // NB*TOT
  float* sel_s = scores + (size_t)NB * TOT;          // NB*KALL
  float* sel_b = sel_s + (size_t)NB * KALL;          // NB*KALL*4

  // Level 0: M=12544 (112x112), 784 tiles; Level 1: M=3136, 196; Level 2: 784, 49
  {
    dim3 g((784 + 3) / 4, NB);
    head_wmma_kernel<12544, 0><<<g, 128, 0, stream>>>(
        reg_feat[0], cls_feat[0], Wconf, Wcls, Wreg, head);
  }
  {
    dim3 g((196 + 3) / 4, NB);
    head_wmma_kernel<3136, 12544><<<g, 128, 0, stream>>>(
        reg_feat[1], cls_feat[1], Wconf + 256, Wcls + 256, Wreg + 1024, head);
  }
  {
    dim3 g((49 + 3) / 4, NB);
    head_wmma_kernel<784, 15680><<<g, 128, 0, stream>>>(
        reg_feat[2], cls_feat[2], Wconf + 512, Wcls + 512, Wreg + 2048, head);
  }

  int total = NB * TOT;
  score_kernel<<<(total + 255) / 256, 256, 0, stream>>>(head, bconf, bcls,
                                                        scores);
  topk_kernel<<<dim3(3, NB), 256, 0, stream>>>(scores, head, breg, sel_s,
                                               sel_b);
  sort_nms_kernel<<<NB, 1024, 0, stream>>>(sel_s, sel_b, (float*)d_out);
}